// MultiBoxLoss_58926951301823
// MI455X (gfx1250) — compile-verified
//
#include <hip/hip_runtime.h>
#include <hip/hip_bf16.h>
#include <stdint.h>

#define BATCH   512
#define NANCH   8732
#define NCHUNK4 (NANCH / 4)   // 2183 float4 chunks per row (exact: 2183*4 == 8732)

__device__ __forceinline__ float smoothL1(float d) {
  float ad = fabsf(d);
  return ad < 1.0f ? 0.5f * d * d : ad - 0.5f;
}

// ---------------------------------------------------------------------------
// Kernel 1: localization loss (masked smooth-L1 sum) per batch row.
// One 256-thread block per row; ~2% of anchors are positive so loads are
// guarded behind the label test.
// ---------------------------------------------------------------------------
__global__ void __launch_bounds__(256) loc_loss_kernel(
    const float* __restrict__ ploc, const float* __restrict__ gloc,
    const int* __restrict__ glabel, const float* __restrict__ dboxes,
    float* __restrict__ gLocLoss)
{
  __shared__ float sred[256];
  const int b   = blockIdx.x;
  const int tid = threadIdx.x;

  const float*  pl0 = ploc + ((size_t)b * 4 + 0) * NANCH;
  const float*  pl1 = ploc + ((size_t)b * 4 + 1) * NANCH;
  const float*  pl2 = ploc + ((size_t)b * 4 + 2) * NANCH;
  const float*  pl3 = ploc + ((size_t)b * 4 + 3) * NANCH;
  const float4* gl  = (const float4*)(gloc + (size_t)b * NANCH * 4);
  const int*    lab = glabel + (size_t)b * NANCH;
  const float4* db  = (const float4*)dboxes;

  float acc = 0.f;
  for (int n = tid; n < NANCH; n += 256) {
    if (lab[n] > 0) {
      float4 d = db[n];
      float4 g = gl[n];
      // point_form: dxy = xy - wh/2 (cols 0,1), dwh = xy + wh/2 (cols 2,3)
      float dxy0 = d.x - 0.5f * d.z, dxy1 = d.y - 0.5f * d.w;
      float dwh0 = d.x + 0.5f * d.z, dwh1 = d.y + 0.5f * d.w;
      float df0 = pl0[n] - (g.x - dxy0) / dwh0;
      float df1 = pl1[n] - (g.y - dxy1) / dwh1;
      float df2 = pl2[n] - logf(g.z / dwh0);
      float df3 = pl3[n] - logf(g.w / dwh1);
      acc += smoothL1(df0) + smoothL1(df1) + smoothL1(df2) + smoothL1(df3);
    }
  }
  sred[tid] = acc;
  __syncthreads();
  for (int s = 128; s > 0; s >>= 1) {
    if (tid < s) sred[tid] += sred[tid + s];
    __syncthreads();
  }
  if (tid == 0) gLocLoss[b] = sred[0];
}

// ---------------------------------------------------------------------------
// Kernel 2: confidence loss + hard-negative mining per batch row.
// Stages pconf rows into LDS with CDNA5 GLOBAL_LOAD_ASYNC_TO_LDS_B128,
// computes con_neg in LDS, then does an exact 4-pass radix select (K-th
// largest) entirely in LDS. All con_neg >= 0 so uint bit order == float order.
// ---------------------------------------------------------------------------
__global__ void __launch_bounds__(256) conf_select_kernel(
    const float* __restrict__ pconf, const int* __restrict__ glabel,
    const float* __restrict__ gLocLoss, float* __restrict__ rowContrib)
{
  __shared__ __align__(16) float sP0[NANCH];
  __shared__ __align__(16) float sP1[NANCH];
  __shared__ __align__(16) float sCon[NANCH];
  __shared__ unsigned int hist[256];
  __shared__ float sredF[256];
  __shared__ int   sredI[256];
  __shared__ unsigned int sPrefix;
  __shared__ int sRemaining;

  const int b   = blockIdx.x;
  const int tid = threadIdx.x;
  const float* p0g = pconf + ((size_t)b * 2 + 0) * NANCH;
  const float* p1g = pconf + ((size_t)b * 2 + 1) * NANCH;
  const int*   lab = glabel + (size_t)b * NANCH;

  // ---- async copy both pconf rows into LDS (tracked by ASYNCcnt) ----
  {
    unsigned int l0 = (unsigned int)(uintptr_t)&sP0[0];
    unsigned int l1 = (unsigned int)(uintptr_t)&sP1[0];
    for (int c = tid; c < NCHUNK4; c += 256) {
      const float* g0 = p0g + c * 4;
      const float* g1 = p1g + c * 4;
      unsigned int d0 = l0 + (unsigned int)(c * 16);
      unsigned int d1 = l1 + (unsigned int)(c * 16);
      asm volatile("global_load_async_to_lds_b128 %0, %1, off"
                   :: "v"(d0), "v"(g0) : "memory");
      asm volatile("global_load_async_to_lds_b128 %0, %1, off"
                   :: "v"(d1), "v"(g1) : "memory");
    }
    asm volatile("s_wait_asynccnt 0" ::: "memory");
  }
  __syncthreads();

  // ---- per-anchor 2-class log-softmax loss; con_neg into LDS ----
  int   posCnt   = 0;
  float posCloss = 0.f;
  for (int n = tid; n < NANCH; n += 256) {
    float p0 = sP0[n], p1 = sP1[n];
    float m   = fmaxf(p0, p1);
    float lse = m + logf(expf(p0 - m) + expf(p1 - m));
    int   lv  = lab[n];
    float closs = lse - (lv > 0 ? p1 : p0);   // -log_softmax at glabel
    bool  mask  = lv > 0;
    sCon[n] = mask ? 0.f : closs;
    if (mask) { posCnt++; posCloss += closs; }
  }
  sredI[tid] = posCnt;
  sredF[tid] = posCloss;
  __syncthreads();
  for (int s = 128; s > 0; s >>= 1) {
    if (tid < s) { sredI[tid] += sredI[tid + s]; sredF[tid] += sredF[tid + s]; }
    __syncthreads();
  }
  const int   pos         = sredI[0];
  const float posClossSum = sredF[0];
  __syncthreads();

  const int K = min(3 * pos, NANCH);
  if (tid == 0) { sPrefix = 0u; sRemaining = K; }
  __syncthreads();

  // ---- 4-pass radix select: K-th largest of sCon ----
  for (int pass = 0; pass < 4; ++pass) {
    const int shift = 24 - 8 * pass;
    hist[tid] = 0u;
    __syncthreads();
    unsigned int pfx = sPrefix;
    for (int n = tid; n < NANCH; n += 256) {
      unsigned int bits = __float_as_uint(sCon[n]);
      bool match = (pass == 0) || ((bits >> (shift + 8)) == (pfx >> (shift + 8)));
      if (match) atomicAdd(&hist[(bits >> shift) & 255u], 1u);
    }
    __syncthreads();
    if (tid == 0) {
      int rem = sRemaining;
      int cum = 0, bin = 255;
      for (; bin > 0; --bin) {            // scan high bin -> low
        int h = (int)hist[bin];
        if (cum + h >= rem) break;
        cum += h;
      }
      sPrefix    = pfx | ((unsigned int)bin << shift);
      sRemaining = rem - cum;             // ties to take at this digit
    }
    __syncthreads();
  }

  const unsigned int tbits   = sPrefix;
  const int          takeAtT = sRemaining;
  const float        tval    = __uint_as_float(tbits);

  // ---- sum of selected negatives: strict-greater sum + tie contribution ----
  float sgt = 0.f;
  for (int n = tid; n < NANCH; n += 256) {
    unsigned int bits = __float_as_uint(sCon[n]);
    if (bits > tbits) sgt += sCon[n];
  }
  __syncthreads();
  sredF[tid] = sgt;
  __syncthreads();
  for (int s = 128; s > 0; s >>= 1) {
    if (tid < s) sredF[tid] += sredF[tid + s];
    __syncthreads();
  }

  if (tid == 0) {
    float negSum = sredF[0] + (takeAtT > 0 ? (float)takeAtT * tval : 0.f);
    float total  = gLocLoss[b] + posClossSum + negSum;
    rowContrib[b] = (pos > 0) ? total / fmaxf((float)pos, 1e-6f) : 0.f;
  }
}

// ---------------------------------------------------------------------------
// Kernel 3: mean over the 512 per-row contributions via chained
// v_wmma_f32_16x16x4_f32 with an all-ones B matrix (full f32 precision).
// A-matrix 16x4 layout (ISA 7.12.2): lanes 0-15 hold K=0,1 in VGPR0,1;
// lanes 16-31 hold K=2,3.  D[m][n] = rowsum_m for every n; column 0 sits in
// lanes 0 (M=0..7) and 16 (M=8..15) across the 8 accumulator VGPRs.
// ---------------------------------------------------------------------------
typedef float v2f __attribute__((ext_vector_type(2)));
typedef float v8f __attribute__((ext_vector_type(8)));

__global__ void __launch_bounds__(32) reduce_mean_wmma(
    const float* __restrict__ rowContrib, float* __restrict__ out)
{
  const int lane  = threadIdx.x;
  const int m     = lane & 15;
  const int kbase = (lane >> 4) << 1;   // 0 for lanes 0-15, 2 for lanes 16-31

  v8f acc = {0.f, 0.f, 0.f, 0.f, 0.f, 0.f, 0.f, 0.f};
  v2f ones; ones[0] = 1.f; ones[1] = 1.f;

  for (int c = 0; c < 8; ++c) {          // 8 chunks of 64 values = 512
    v2f a;
    a[0] = rowContrib[c * 64 + m * 4 + kbase + 0];
    a[1] = rowContrib[c * 64 + m * 4 + kbase + 1];
    acc = __builtin_amdgcn_wmma_f32_16x16x4_f32(false, a, false, ones,
                                                (short)0, acc, false, false);
  }
  float s  = acc[0] + acc[1] + acc[2] + acc[3] + acc[4] + acc[5] + acc[6] + acc[7];
  float hi = __shfl(s, 16, 32);          // lane 16 holds rowsums M=8..15
  if (lane == 0) out[0] = (s + hi) / (float)BATCH;
}

// ---------------------------------------------------------------------------
extern "C" void kernel_launch(void* const* d_in, const int* in_sizes, int n_in,
                              void* d_out, int out_size, void* d_ws, size_t ws_size,
                              hipStream_t stream) {
  const float* ploc   = (const float*)d_in[0];   // (B,4,N)
  const float* pconf  = (const float*)d_in[1];   // (B,2,N)
  const float* gloc   = (const float*)d_in[2];   // (B,N,4)
  const int*   glabel = (const int*)  d_in[3];   // (B,N)
  const float* dboxes = (const float*)d_in[4];   // (N,4)
  float* out = (float*)d_out;

  float* gLocLoss   = (float*)d_ws;              // [BATCH]
  float* rowContrib = gLocLoss + BATCH;          // [BATCH]

  loc_loss_kernel   <<<BATCH, 256, 0, stream>>>(ploc, gloc, glabel, dboxes, gLocLoss);
  conf_select_kernel<<<BATCH, 256, 0, stream>>>(pconf, glabel, gLocLoss, rowContrib);
  reduce_mean_wmma  <<<1,     32,  0, stream>>>(rowContrib, out);
}